// MoELayer_90984587198472
// MI455X (gfx1250) — compile-verified
//
#include <hip/hip_runtime.h>
#include <math.h>

// ---------------------------------------------------------------------------
// MoE (buggy-gather-faithful) for MI455X / gfx1250.
//
// Shapes: B=4 S=2048 D=1024 O=1024 E=8 K=2.  Only feature columns o=0,1 of
// each expert matter (reference gathers feature dim with slot index j), so:
//   GEMM1: logits  = x[8192,1024] @ gate_W^T[1024,16]   (8 real + 8 pad cols)
//   GEMM2: colvals = x[8192,1024] @ Wcols^T [1024,16]   (cols n = 2e+j)
// both done with V_WMMA_F32_16X16X4_F32 (exact f32, no precision loss for
// the sigmoid top-k).  Memory-bound: ~64 MB HBM traffic -> ~2.8us floor.
// ---------------------------------------------------------------------------

typedef __attribute__((ext_vector_type(2))) float v2f;
typedef __attribute__((ext_vector_type(4))) float f4;
typedef __attribute__((ext_vector_type(8))) float v8f;

constexpr int Bb = 4, Ss = 2048, Dd = 1024, Oo = 1024, Ee = 8;
constexpr int TOKENS = Bb * Ss;            // 8192
constexpr int CHUNKS = Dd / 4;             // 256 K-steps of 4
constexpr int FRAG_FLOATS = 64;            // 32 lanes * 2 floats per B-frag chunk
constexpr int WAVES = 4;                   // waves per block (1 tile of 16 tokens each)

// ---------------------------------------------------------------------------
// Pack B operands into WMMA fragment order, one 64-float fragment per K-chunk.
// Fragment element (chunk c, lane l, vgpr r) = Bmat[k][n] with
//   n = l & 15,  k = (l>>4)*2 + r   (f32 16x16x4 B layout, C/D-analog split)
// ws[0        .. 16383] : gate_W  (cols 8..15 zero-padded)
// ws[16384 .. 32767]    : expert columns, n = 2*e + j  ->  W_experts[e, j, :]
// ---------------------------------------------------------------------------
__global__ void moe_pack_b(const float* __restrict__ gate_W,
                           const float* __restrict__ W_experts,
                           float* __restrict__ ws) {
    int idx = blockIdx.x * blockDim.x + threadIdx.x;      // 0..32767
    if (idx >= 2 * CHUNKS * FRAG_FLOATS) return;
    int which = idx >> 14;                                // 0 = gate, 1 = cols
    int e2    = idx & 16383;                              // c*64 + l*2 + r
    int c = e2 >> 6;
    int l = (e2 >> 1) & 31;
    int r = e2 & 1;
    int n = l & 15;
    int k = ((l >> 4) << 1) + r;
    int d = c * 4 + k;
    float v;
    if (which == 0) {
        v = (n < Ee) ? gate_W[n * Dd + d] : 0.0f;
    } else {
        int e = n >> 1, j = n & 1;
        v = W_experts[((size_t)(e * Oo + j)) * Dd + d];
    }
    ws[(size_t)which * (CHUNKS * FRAG_FLOATS) + e2] = v;
}

// ---------------------------------------------------------------------------
// Main fused kernel: one wave32 per 16-token tile.
// ---------------------------------------------------------------------------
__global__ void __launch_bounds__(WAVES * 32)
moe_main(const float* __restrict__ x,
         const float* __restrict__ b_experts,
         const float* __restrict__ gate_b,
         const float* __restrict__ expert_biases,
         const float* __restrict__ ws,
         float* __restrict__ out,
         int* __restrict__ out_idx) {
    __shared__ float ldsG[WAVES][16][16];   // gate logits   [m][e]
    __shared__ float ldsW[WAVES][16][16];   // expert cols   [m][2e+j]
    __shared__ float ldsC[WAVES][16];       // combined value per token

    const int wave   = threadIdx.x >> 5;
    const int lane   = threadIdx.x & 31;
    const int tile   = blockIdx.x * WAVES + wave;
    const int token0 = tile * 16;

    const int n  = lane & 15;   // A: row M / B: col N
    const int kh = lane >> 4;   // K half-select (ISA 16x4 f32 A layout)

    // A fragment: lane holds x[token0+n, c*4 + kh*2 + {0,1}] -> b64 loads
    const float* __restrict__ xrow  = x + (size_t)(token0 + n) * Dd + kh * 2;
    const float* __restrict__ gfrag = ws + lane * 2;
    const float* __restrict__ wfrag = ws + (size_t)CHUNKS * FRAG_FLOATS + lane * 2;

    v8f cg = {};   // 16x16 gate-logit accumulator
    v8f cw = {};   // 16x16 expert-column accumulator

    #pragma unroll 4
    for (int c = 0; c < CHUNKS; ++c) {
        v2f a  = *(const v2f*)(xrow  + (size_t)c * 4);
        v2f bg = *(const v2f*)(gfrag + (size_t)c * FRAG_FLOATS);
        v2f bw = *(const v2f*)(wfrag + (size_t)c * FRAG_FLOATS);
        // D = A*B + C  (neg_a, A, neg_b, B, c_mod, C, reuse_a, reuse_b)
        cg = __builtin_amdgcn_wmma_f32_16x16x4_f32(false, a, false, bg,
                                                   (short)0, cg, false, false);
        cw = __builtin_amdgcn_wmma_f32_16x16x4_f32(false, a, false, bw,
                                                   (short)0, cw, false, false);
    }

    // C/D layout: vgpr r, lanes 0-15 -> M=r, lanes 16-31 -> M=r+8; N = lane&15
    #pragma unroll
    for (int r = 0; r < 8; ++r) {
        ldsG[wave][r + 8 * kh][n] = cg[r];
        ldsW[wave][r + 8 * kh][n] = cw[r];
    }
    __syncthreads();

    // ---- scalar epilogue: sigmoid gating, top-2, combine (lanes 0-15) ----
    if (lane < 16) {
        const int m = lane;
        float p0 = -1.0f, p1 = -1.0f;
        int   i0 = 0,     i1 = 0;
        #pragma unroll
        for (int e = 0; e < Ee; ++e) {
            float logit = ldsG[wave][m][e] + gate_b[e] + expert_biases[e];
            float p = 1.0f / (1.0f + __expf(-logit));
            if (p > p0)      { p1 = p0; i1 = i0; p0 = p; i0 = e; }
            else if (p > p1) { p1 = p;  i1 = e; }
        }
        float inv = 1.0f / (p0 + p1);
        // slot j selects feature column j of its expert (faithful to the bug)
        float sel0 = ldsW[wave][m][i0 * 2 + 0] + b_experts[i0 * Oo + 0];
        float sel1 = ldsW[wave][m][i1 * 2 + 1] + b_experts[i1 * Oo + 1];
        ldsC[wave][m] = sel0 * (p0 * inv) + sel1 * (p1 * inv);
        out_idx[(size_t)(token0 + m) * 2 + 0] = i0;
        out_idx[(size_t)(token0 + m) * 2 + 1] = i1;
    }
    __syncthreads();

    // ---- broadcast combined value across O=1024, b128 coalesced stores ----
    #pragma unroll 2
    for (int tt = 0; tt < 16; ++tt) {
        float cv = ldsC[wave][tt];
        f4 v = {cv, cv, cv, cv};
        f4* __restrict__ dst = (f4*)(out + (size_t)(token0 + tt) * Oo);
        #pragma unroll
        for (int i = 0; i < 8; ++i) dst[i * 32 + lane] = v;
    }
}

// ---------------------------------------------------------------------------
extern "C" void kernel_launch(void* const* d_in, const int* in_sizes, int n_in,
                              void* d_out, int out_size, void* d_ws, size_t ws_size,
                              hipStream_t stream) {
    const float* x             = (const float*)d_in[0];
    const float* W_experts     = (const float*)d_in[1];
    const float* b_experts     = (const float*)d_in[2];
    const float* gate_W        = (const float*)d_in[3];
    const float* gate_b        = (const float*)d_in[4];
    const float* expert_biases = (const float*)d_in[5];

    float* out     = (float*)d_out;                       // [B,S,O] f32
    int*   out_idx = (int*)(out + (size_t)TOKENS * Oo);   // [B,S,2] i32, concat

    float* ws = (float*)d_ws;                             // 128 KB B-fragments

    moe_pack_b<<<(2 * CHUNKS * FRAG_FLOATS + 255) / 256, 256, 0, stream>>>(
        gate_W, W_experts, ws);

    moe_main<<<TOKENS / 16 / WAVES, WAVES * 32, 0, stream>>>(
        x, b_experts, gate_b, expert_biases, ws, out, out_idx);
}